// SolarRingLayer_74062416053186
// MI455X (gfx1250) — compile-verified
//
#include <hip/hip_runtime.h>
#include <math.h>

typedef __attribute__((ext_vector_type(2))) float v2f;
typedef __attribute__((ext_vector_type(4))) float v4f;
typedef __attribute__((ext_vector_type(8))) float v8f;
typedef __attribute__((ext_vector_type(4))) int   v4i;

#define DD 1024
#define NN 32768

#define BK    32              // K-chunk staged in LDS
#define LDSTR 36              // row stride in floats (32 + 4 pad -> conflict-free ds_load_b64)
#define TILEF (128 * LDSTR)   // floats per (x or w) tile buffer
#define NCH   (DD / BK)       // 32 K-chunks

// d_out layout (float element offsets), outputs concatenated in return order:
// x_out (N*D), role_logits (N*6), spawn_logit (N), write_vec (N*D), verb_gate (N)
#define OFF_XOUT  ((size_t)0)
#define OFF_ROLE  ((size_t)NN * DD)
#define OFF_SPAWN (OFF_ROLE + (size_t)6 * NN)
#define OFF_WRITE (OFF_SPAWN + (size_t)NN)
#define OFF_VG    (OFF_WRITE + (size_t)NN * DD)

// ---- CDNA5 async direct-to-LDS path (guarded; manual copy fallback) -------
#if defined(__has_builtin)
#  if __has_builtin(__builtin_amdgcn_global_load_async_to_lds_b128)
#    define HAS_ASYNC_LDS 1
#  endif
#endif

#define AS1 __attribute__((address_space(1)))
#define AS3 __attribute__((address_space(3)))

__device__ __forceinline__ void copy16_to_lds(const float* gsrc, float* ldst) {
#if defined(HAS_ASYNC_LDS)
    __builtin_amdgcn_global_load_async_to_lds_b128(
        (AS1 v4i*)gsrc, (AS3 v4i*)ldst, 0, 0);
#else
    *(v4f*)ldst = *(const v4f*)gsrc;
#endif
}

#if defined(HAS_ASYNC_LDS)
#  if __has_builtin(__builtin_amdgcn_s_wait_asynccnt)
#    define WAIT_ASYNC(n) __builtin_amdgcn_s_wait_asynccnt(n)
#  else
#    define WAIT_ASYNC(n) asm volatile("s_wait_asynccnt " #n ::: "memory")
#  endif
#else
#  define WAIT_ASYNC(n) do {} while (0)
#endif

__device__ __forceinline__ float sigmoidf_(float v) {
    return 1.0f / (1.0f + expf(-v));
}

// ---------------------------------------------------------------------------
// Kernel 1: small heads. One wave32 per row: role_logits (6), spawn_logit,
// verb_gate = sigmoid(x . W_vg + b_vg). Also handles role==CONJ (3):
// write_vec = x.
// ---------------------------------------------------------------------------
__global__ void __launch_bounds__(256) small_heads_kernel(
    const float* __restrict__ x, const int* __restrict__ role,
    const float* __restrict__ W_role, const float* __restrict__ b_role,
    const float* __restrict__ W_spawn, const float* __restrict__ b_spawn,
    const float* __restrict__ W_vg, const float* __restrict__ b_vg,
    float* __restrict__ out)
{
    const int wave = threadIdx.x >> 5;
    const int lane = threadIdx.x & 31;
    const int row  = blockIdx.x * 8 + wave;
    const float* __restrict__ xr = x + (size_t)row * DD;

    float acc[8];
#pragma unroll
    for (int j = 0; j < 8; ++j) acc[j] = 0.0f;

#pragma unroll
    for (int it = 0; it < 8; ++it) {
        const int k = (it * 32 + lane) * 4;
        v4f xv = *(const v4f*)(xr + k);
#pragma unroll
        for (int j = 0; j < 6; ++j) {
            v4f wv = *(const v4f*)(W_role + j * DD + k);
            acc[j] += xv.x * wv.x + xv.y * wv.y + xv.z * wv.z + xv.w * wv.w;
        }
        v4f sv = *(const v4f*)(W_spawn + k);
        acc[6] += xv.x * sv.x + xv.y * sv.y + xv.z * sv.z + xv.w * sv.w;
        v4f gv = *(const v4f*)(W_vg + k);
        acc[7] += xv.x * gv.x + xv.y * gv.y + xv.z * gv.z + xv.w * gv.w;
    }

#pragma unroll
    for (int off = 16; off >= 1; off >>= 1) {
#pragma unroll
        for (int j = 0; j < 8; ++j)
            acc[j] += __shfl_xor(acc[j], off, 32);
    }

    if (lane == 0) {
#pragma unroll
        for (int j = 0; j < 6; ++j)
            out[OFF_ROLE + (size_t)row * 6 + j] = acc[j] + b_role[j];
        out[OFF_SPAWN + row] = acc[6] + b_spawn[0];
        out[OFF_VG + row]    = sigmoidf_(acc[7] + b_vg[0]);
    }

    if (role[row] == 3) {   // CONJ: write_vec = x
        float* __restrict__ wv = out + OFF_WRITE + (size_t)row * DD;
#pragma unroll
        for (int it = 0; it < 8; ++it) {
            const int k = (it * 32 + lane) * 4;
            *(v4f*)(wv + k) = *(const v4f*)(xr + k);
        }
    }
}

// ---------------------------------------------------------------------------
// Kernel 2: fused GEMM over virtual 5*D column space, f32 WMMA.
// Block tile 128x128; 8 waves = 4 row-groups x 2 col-groups; wave tile 32x64.
// Double-buffered async global->LDS staging (ASYNCcnt) overlaps the DMA of
// chunk c+1 with compute of chunk c. LDS layout per buffer: xs[128][36],
// ws[128][36] (pad 4 -> conflict-free ds_load_b64 fragment reads).
// ---------------------------------------------------------------------------
__global__ void __launch_bounds__(256) fused_gemm_kernel(
    const float* __restrict__ x, const int* __restrict__ role,
    const float* __restrict__ W_subj, const float* __restrict__ b_subj,
    const float* __restrict__ W_obj,  const float* __restrict__ b_obj,
    const float* __restrict__ W_vc,   const float* __restrict__ b_vc,
    const float* __restrict__ W_rot,  const float* __restrict__ b_rot,
    const float* __restrict__ W_og,   const float* __restrict__ b_og,
    float* __restrict__ out)
{
    extern __shared__ float smem[];   // 2 buffers x (xs + ws) = 4 * TILEF floats

    const int NCB = DD / 128;                // 8 column blocks per matrix
    const int cb  = blockIdx.x % (5 * NCB);  // 40 virtual column blocks
    const int rb  = blockIdx.x / (5 * NCB);
    const int mid = cb / NCB;                // 0 subj, 1 obj, 2 vc, 3 rot, 4 og
    const int colBase = (cb % NCB) * 128;
    const int rowBase = rb * 128;

    const int tid  = threadIdx.x;
    const int wave = tid >> 5;
    const int lane = tid & 31;
    const int l15  = lane & 15;
    const int lhi  = lane >> 4;
    const int rg   = wave >> 1;              // row group 0..3 (32 rows)
    const int cg   = wave & 1;               // col group 0..1 (64 cols)

    const float* W;  const float* bias;
    switch (mid) {
        case 0:  W = W_subj; bias = b_subj; break;
        case 1:  W = W_obj;  bias = b_obj;  break;
        case 2:  W = W_vc;   bias = b_vc;   break;
        case 3:  W = W_rot;  bias = b_rot;  break;
        default: W = W_og;   bias = b_og;   break;
    }

    // Per-thread staging coordinates (4 b128 transfers per tile per thread).
    // li = tid + 256*i -> row = li/8, kq = (li%8)*4.
    auto issue_chunk = [&](int buf, int kb) {
        float* xd = smem + buf * (2 * TILEF);
        float* wd = xd + TILEF;
#pragma unroll
        for (int i = 0; i < 4; ++i) {
            const int li = tid + 256 * i;
            const int r  = li >> 3;
            const int kq = (li & 7) << 2;
            copy16_to_lds(x + (size_t)(rowBase + r) * DD + kb + kq, &xd[r * LDSTR + kq]);
            copy16_to_lds(W + (size_t)(colBase + r) * DD + kb + kq, &wd[r * LDSTR + kq]);
        }
    };

    v8f acc[2][4];
#pragma unroll
    for (int a = 0; a < 2; ++a)
#pragma unroll
        for (int j = 0; j < 4; ++j)
            acc[a][j] = (v8f){0.f,0.f,0.f,0.f,0.f,0.f,0.f,0.f};

    // Prime the pipeline: chunks 0 and 1 in flight (8 async ops each per wave).
    issue_chunk(0, 0);
    issue_chunk(1, BK);

    for (int c = 0; c < NCH; ++c) {
        const int cur = c & 1;
        if (c < NCH - 1) { WAIT_ASYNC(8); }   // chunk c landed (in-order), c+1 may fly
        else             { WAIT_ASYNC(0); }
        __syncthreads();                       // all waves' chunk-c data visible

        const float* xsb = smem + cur * (2 * TILEF);
        const float* wsb = xsb + TILEF;

#pragma unroll
        for (int kk = 0; kk < BK; kk += 4) {
            const float* ap = &xsb[(rg * 32 + l15) * LDSTR + kk + 2 * lhi];
            v2f a0 = *(const v2f*)ap;
            v2f a1 = *(const v2f*)(ap + 16 * LDSTR);
#pragma unroll
            for (int j = 0; j < 4; ++j) {
                v2f b = *(const v2f*)(&wsb[(cg * 64 + j * 16 + l15) * LDSTR + kk + 2 * lhi]);
                acc[0][j] = __builtin_amdgcn_wmma_f32_16x16x4_f32(false, a0, false, b, (short)0, acc[0][j], false, false);
                acc[1][j] = __builtin_amdgcn_wmma_f32_16x16x4_f32(false, a1, false, b, (short)0, acc[1][j], false, false);
            }
        }

        __syncthreads();                       // all waves done reading buffer cur
        if (c + 2 < NCH) issue_chunk(cur, (c + 2) * BK);
    }

    // Epilogue. C/D layout: VGPR i holds (M = i + 8*lhi, N = l15).
    const float* __restrict__ vg  = out + OFF_VG;
    float* __restrict__ wvout = out + OFF_WRITE;
    float* __restrict__ xout  = out + OFF_XOUT;

#pragma unroll
    for (int ar = 0; ar < 2; ++ar) {
#pragma unroll
        for (int j = 0; j < 4; ++j) {
            const int col = colBase + cg * 64 + j * 16 + l15;
            const float bcol = bias[col];
#pragma unroll
            for (int i = 0; i < 8; ++i) {
                const int row = rowBase + rg * 32 + ar * 16 + 8 * lhi + i;
                const float v = acc[ar][j][i] + bcol;
                const size_t idx = (size_t)row * DD + col;
                if (mid == 4) {
                    const float g  = sigmoidf_(v);
                    const float xv = x[idx];
                    xout[idx] = xv + g * xv;          // t, LayerNorm applied later
                } else {
                    const int r = role[row];
                    if (mid == 2)      { if (r == 2) wvout[idx] = vg[row] * v; }
                    else if (mid == 3) { if (r >= 4) wvout[idx] = v; }
                    else               { if (r == mid) wvout[idx] = v; }
                }
            }
        }
    }
}

// ---------------------------------------------------------------------------
// Kernel 3: in-place LayerNorm of t (stored in x_out region). One block/row.
// ---------------------------------------------------------------------------
__global__ void __launch_bounds__(256) layernorm_kernel(
    const float* __restrict__ gamma, const float* __restrict__ beta,
    float* __restrict__ out)
{
    __shared__ float s_sum[8];
    __shared__ float s_sq[8];

    const int row = blockIdx.x;
    float* __restrict__ t = out + OFF_XOUT + (size_t)row * DD;
    const int tid  = threadIdx.x;
    const int wave = tid >> 5;
    const int lane = tid & 31;

    v4f v = *(const v4f*)(t + tid * 4);
    float s = v.x + v.y + v.z + v.w;
    float q = v.x * v.x + v.y * v.y + v.z * v.z + v.w * v.w;

#pragma unroll
    for (int off = 16; off >= 1; off >>= 1) {
        s += __shfl_xor(s, off, 32);
        q += __shfl_xor(q, off, 32);
    }
    if (lane == 0) { s_sum[wave] = s; s_sq[wave] = q; }
    __syncthreads();

    float S = 0.f, Q = 0.f;
#pragma unroll
    for (int w = 0; w < 8; ++w) { S += s_sum[w]; Q += s_sq[w]; }

    const float mu  = S * (1.0f / DD);
    const float var = Q * (1.0f / DD) - mu * mu;
    const float inv = rsqrtf(var + 1e-5f);

    v4f g = *(const v4f*)(gamma + tid * 4);
    v4f b = *(const v4f*)(beta + tid * 4);
    v4f o;
    o.x = (v.x - mu) * inv * g.x + b.x;
    o.y = (v.y - mu) * inv * g.y + b.y;
    o.z = (v.z - mu) * inv * g.z + b.z;
    o.w = (v.w - mu) * inv * g.w + b.w;
    *(v4f*)(t + tid * 4) = o;
}

// ---------------------------------------------------------------------------
extern "C" void kernel_launch(void* const* d_in, const int* in_sizes, int n_in,
                              void* d_out, int out_size, void* d_ws, size_t ws_size,
                              hipStream_t stream) {
    const float* x       = (const float*)d_in[0];
    const int*   role    = (const int*)  d_in[1];
    const float* W_role  = (const float*)d_in[2];
    const float* b_role  = (const float*)d_in[3];
    const float* W_spawn = (const float*)d_in[4];
    const float* b_spawn = (const float*)d_in[5];
    const float* W_subj  = (const float*)d_in[6];
    const float* b_subj  = (const float*)d_in[7];
    const float* W_obj   = (const float*)d_in[8];
    const float* b_obj   = (const float*)d_in[9];
    const float* W_vg    = (const float*)d_in[10];
    const float* b_vg    = (const float*)d_in[11];
    const float* W_vc    = (const float*)d_in[12];
    const float* b_vc    = (const float*)d_in[13];
    const float* W_rot   = (const float*)d_in[14];
    const float* b_rot   = (const float*)d_in[15];
    const float* W_og    = (const float*)d_in[16];
    const float* b_og    = (const float*)d_in[17];
    const float* gamma   = (const float*)d_in[18];
    const float* beta    = (const float*)d_in[19];
    float* out = (float*)d_out;

    small_heads_kernel<<<NN / 8, 256, 0, stream>>>(
        x, role, W_role, b_role, W_spawn, b_spawn, W_vg, b_vg, out);

    const size_t smem_bytes = (size_t)4 * TILEF * sizeof(float);  // 73,728 B
    fused_gemm_kernel<<<(NN / 128) * (5 * (DD / 128)), 256, smem_bytes, stream>>>(
        x, role, W_subj, b_subj, W_obj, b_obj, W_vc, b_vc,
        W_rot, b_rot, W_og, b_og, out);

    layernorm_kernel<<<NN, 256, 0, stream>>>(gamma, beta, out);
}